// PhoneticLatentSpace_15075335209401
// MI455X (gfx1250) — compile-verified
//
#include <hip/hip_runtime.h>
#include <math.h>

// ---------------- problem constants (match reference) ----------------
#define VOCAB   256
#define B_ROWS  65536
#define S_LEN   128
#define P_ROWS  512
#define D_DIM   1024
#define EPSF    1e-8f

typedef float v2f __attribute__((ext_vector_type(2)));
typedef float v8f __attribute__((ext_vector_type(8)));

// low 32 bits of a generic pointer to LDS == byte offset within the
// workgroup's LDS allocation
__device__ __forceinline__ unsigned lds_off(const void* p) {
    return (unsigned)(size_t)p;
}

// CDNA5 async copy: global -> LDS, 16 bytes per lane, tracked by ASYNCcnt.
// GVS mode: mem_addr = SGPR64(base) + VGPR_I32(byte offset)
__device__ __forceinline__ void async_ld_b128(unsigned lds_byte_off, int glb_byte_off,
                                              const float* base) {
    asm volatile("global_load_async_to_lds_b128 %0, %1, %2"
                 :: "v"(lds_byte_off), "v"(glb_byte_off), "s"(base)
                 : "memory");
}

// ---------------------------------------------------------------------
// Kernel 0: zero the atomically-accumulated output regions
// ---------------------------------------------------------------------
__global__ void zero_kernel(float* __restrict__ cooc, float* __restrict__ usage,
                            float* __restrict__ ctx) {
    int i = blockIdx.x * 256 + threadIdx.x;
    if (i < VOCAB * VOCAB)   cooc[i]  = 0.0f;
    if (i < P_ROWS)          usage[i] = 0.0f;
    if (i < P_ROWS * D_DIM)  ctx[i]   = 0.0f;
}

// ---------------------------------------------------------------------
// Kernel 1: bigram co-occurrence histogram (native f32 global atomics)
// ---------------------------------------------------------------------
__global__ void cooc_kernel(const int* __restrict__ phon, float* __restrict__ cooc) {
    int gid = blockIdx.x * 256 + threadIdx.x;
    const int total = B_ROWS * (S_LEN - 1);
    if (gid >= total) return;
    int row = gid / (S_LEN - 1);
    int col = gid - row * (S_LEN - 1);
    int p1 = phon[row * S_LEN + col];
    int p2 = phon[row * S_LEN + col + 1];
    if (p1 < VOCAB && p2 < VOCAB) {
        int c1 = min(max(p1, 0), VOCAB - 1);
        int c2 = min(max(p2, 0), VOCAB - 1);
        unsafeAtomicAdd(&cooc[c1 * VOCAB + c2], 1.0f);
    }
}

// ---------------------------------------------------------------------
// Kernel 2: inverse row norms for hidden (B rows) and patterns (P rows)
// one wave32 per row; float4 loads; shfl_xor allreduce
// ---------------------------------------------------------------------
__global__ void norms_kernel(const float* __restrict__ hid, const float* __restrict__ pat,
                             float* __restrict__ invH, float* __restrict__ invP) {
    int wid  = (blockIdx.x * 256 + threadIdx.x) >> 5;
    int lane = threadIdx.x & 31;
    const int totalRows = B_ROWS + P_ROWS;
    if (wid >= totalRows) return;
    const float* src = (wid < B_ROWS) ? (hid + (size_t)wid * D_DIM)
                                      : (pat + (size_t)(wid - B_ROWS) * D_DIM);
    const float4* v = (const float4*)src;
    float s = 0.0f;
#pragma unroll
    for (int i = 0; i < 8; ++i) {            // 8 * 32 lanes * 4 = 1024 floats
        float4 x = v[lane + 32 * i];
        s += x.x * x.x + x.y * x.y + x.z * x.z + x.w * x.w;
    }
#pragma unroll
    for (int off = 16; off > 0; off >>= 1) s += __shfl_xor(s, off, 32);
    if (lane == 0) {
        float inv = 1.0f / fmaxf(sqrtf(s), EPSF);
        if (wid < B_ROWS) invH[wid] = inv;
        else              invP[wid - B_ROWS] = inv;
    }
}

// ---------------------------------------------------------------------
// Kernel 3: similarity GEMM with V_WMMA_F32_16X16X4_F32 +
//           double-buffered GLOBAL_LOAD_ASYNC_TO_LDS_B128 staging.
//
//   C block per workgroup: 64 (M) x 128 (N); 8 waves as 4(M) x 2(N),
//   each wave owns a 16x64 tile = 4 x v8f accumulators.
//   K slabs of 32 stream into LDS buf[i&1] via async loads issued one
//   slab ahead; each wave issues 6 b128 async loads per slab, so
//   s_wait_asynccnt 6 after issuing slab i+1 guarantees slab i landed
//   (async loads complete in order per wave), then barrier -> compute.
//
//   Row pad 36 floats keeps the 16-lane frag reads conflict-free
//   (stride 36, 9M mod 16 distinct) and 16B alignment for b128 stores.
//
// WMMA operand layout (ISA 7.12.2):
//   A 16x4 : lane<16 -> A[M=lane][K={0,1}] in {v0,v1}; lane>=16 -> K={2,3}
//   B 4x16 : mirrored over N (lane<16 -> B[K={0,1}][N=lane], lane>=16 -> K={2,3})
//   C 16x16: vgpr r -> row r + 8*(lane>=16), col lane&15
// ---------------------------------------------------------------------
__global__ __launch_bounds__(256)
void gemm_wmma_kernel(const float* __restrict__ hid, const float* __restrict__ pat,
                      const float* __restrict__ invH, const float* __restrict__ invP,
                      float* __restrict__ sim) {
    __shared__ __align__(16) float As[2][64][36];
    __shared__ __align__(16) float Bs[2][128][36];

    const int t     = threadIdx.x;
    const int wave  = t >> 5;
    const int lane  = t & 31;
    const int half  = lane >> 4;
    const int l16   = lane & 15;
    const int mBase = (wave & 3) * 16;   // wave's M offset inside block tile
    const int nBase = (wave >> 2) * 64;  // wave's N offset inside block tile
    const int blockM = blockIdx.x * 64;
    const int blockN = blockIdx.y * 128;

    // per-thread staging slot: row group rowBase+32r, 4-float column colF
    const int rowBase = t >> 3;          // 0..31
    const int colF    = (t & 7) * 4;     // float index 0..28 (16B aligned)

    // global byte offsets (without k-slab term); fit in i32
    int aG[2], bG[4];
#pragma unroll
    for (int r = 0; r < 2; ++r)
        aG[r] = ((blockM + rowBase + 32 * r) * D_DIM + colF) * 4;
#pragma unroll
    for (int r = 0; r < 4; ++r)
        bG[r] = ((blockN + rowBase + 32 * r) * D_DIM + colF) * 4;

    // issue one K-slab (32 floats wide) of async loads into LDS buffer `buf`
    auto issue_slab = [&](int k0, int buf) {
        const int kbytes = k0 * 4;
#pragma unroll
        for (int r = 0; r < 2; ++r)
            async_ld_b128(lds_off(&As[buf][rowBase + 32 * r][colF]), aG[r] + kbytes, hid);
#pragma unroll
        for (int r = 0; r < 4; ++r)
            async_ld_b128(lds_off(&Bs[buf][rowBase + 32 * r][colF]), bG[r] + kbytes, pat);
    };

    v8f acc0 = {}, acc1 = {}, acc2 = {}, acc3 = {};

    issue_slab(0, 0);                    // prologue: slab 0 -> buf 0

    const int NSLAB = D_DIM / 32;        // 32
    for (int it = 0; it < NSLAB; ++it) {
        const int cur = it & 1;
        if (it + 1 < NSLAB) {
            issue_slab((it + 1) * 32, cur ^ 1);
            asm volatile("s_wait_asynccnt 0x6" ::: "memory");   // slab `it` landed
        } else {
            asm volatile("s_wait_asynccnt 0x0" ::: "memory");
        }
        __syncthreads();                 // all waves' slab-`it` data visible

        // ---- 8 K-steps of 4; 4 WMMAs per step (one per N sub-tile) ----
#pragma unroll
        for (int kb = 0; kb < 8; ++kb) {
            const int kk = 4 * kb + 2 * half;
            v2f a  = *(const v2f*)&As[cur][mBase + l16][kk];
            v2f b0 = *(const v2f*)&Bs[cur][nBase +  0 + l16][kk];
            v2f b1 = *(const v2f*)&Bs[cur][nBase + 16 + l16][kk];
            v2f b2 = *(const v2f*)&Bs[cur][nBase + 32 + l16][kk];
            v2f b3 = *(const v2f*)&Bs[cur][nBase + 48 + l16][kk];
            acc0 = __builtin_amdgcn_wmma_f32_16x16x4_f32(false, a, false, b0, (short)0, acc0, false, false);
            acc1 = __builtin_amdgcn_wmma_f32_16x16x4_f32(false, a, false, b1, (short)0, acc1, false, false);
            acc2 = __builtin_amdgcn_wmma_f32_16x16x4_f32(false, a, false, b2, (short)0, acc2, false, false);
            acc3 = __builtin_amdgcn_wmma_f32_16x16x4_f32(false, a, false, b3, (short)0, acc3, false, false);
        }
        __syncthreads();                 // done reading buf `cur` before slab it+2 overwrites it
    }

    // ---- epilogue: scale by inverse norms, store 16x64 tile ----
    const int gn0 = blockN + nBase;
    const float pI0 = invP[gn0 +  0 + l16];
    const float pI1 = invP[gn0 + 16 + l16];
    const float pI2 = invP[gn0 + 32 + l16];
    const float pI3 = invP[gn0 + 48 + l16];
#pragma unroll
    for (int r = 0; r < 8; ++r) {
        const int gm = blockM + mBase + r + 8 * half;
        const float hI = invH[gm];
        float* dst = sim + (size_t)gm * P_ROWS + gn0;
        dst[ 0 + l16] = acc0[r] * hI * pI0;
        dst[16 + l16] = acc1[r] * hI * pI1;
        dst[32 + l16] = acc2[r] * hI * pI2;
        dst[48 + l16] = acc3[r] * hI * pI3;
    }
}

// ---------------------------------------------------------------------
// Kernel 4: fused per-row argmax (first-max tie-break like jnp.argmax)
//           + usage_count and context_sum scatter (native f32 atomics)
// one wave32 per row
// ---------------------------------------------------------------------
__global__ void argmax_scatter_kernel(const float* __restrict__ sim,
                                      const float* __restrict__ hid,
                                      float* __restrict__ top, float* __restrict__ usage,
                                      float* __restrict__ ctx) {
    int wid  = (blockIdx.x * 256 + threadIdx.x) >> 5;
    int lane = threadIdx.x & 31;
    if (wid >= B_ROWS) return;

    const float* row = sim + (size_t)wid * P_ROWS;
    float best = -INFINITY;
    int   bidx = 0x7fffffff;
#pragma unroll
    for (int i = 0; i < 16; ++i) {          // ascending idx per lane -> first max wins
        int idx = lane + 32 * i;
        float v = row[idx];
        if (v > best) { best = v; bidx = idx; }
    }
#pragma unroll
    for (int off = 16; off > 0; off >>= 1) { // allreduce, lower index wins ties
        float ov = __shfl_xor(best, off, 32);
        int   oi = __shfl_xor(bidx, off, 32);
        if (ov > best || (ov == best && oi < bidx)) { best = ov; bidx = oi; }
    }
    if (lane == 0) {
        top[wid] = (float)bidx;
        unsafeAtomicAdd(&usage[bidx], 1.0f);
    }
    const float* h = hid + (size_t)wid * D_DIM;
    float*       c = ctx + (size_t)bidx * D_DIM;
#pragma unroll
    for (int i = 0; i < 32; ++i)
        unsafeAtomicAdd(&c[lane + 32 * i], h[lane + 32 * i]);
}

// ---------------------------------------------------------------------
extern "C" void kernel_launch(void* const* d_in, const int* in_sizes, int n_in,
                              void* d_out, int out_size, void* d_ws, size_t ws_size,
                              hipStream_t stream) {
    const int*   phon = (const int*)  d_in[0];   // (B, S) int32
    const float* hid  = (const float*)d_in[1];   // (B, D) f32
    const float* pat  = (const float*)d_in[2];   // (P, D) f32

    float* out   = (float*)d_out;
    float* cooc  = out;                                   // 256*256
    float* sim   = cooc + VOCAB * VOCAB;                  // B*P
    float* top   = sim  + (size_t)B_ROWS * P_ROWS;        // B
    float* usage = top  + B_ROWS;                         // P
    float* ctx   = usage + P_ROWS;                        // P*D

    float* invH = (float*)d_ws;                           // B floats
    float* invP = invH + B_ROWS;                          // P floats

    // 0) zero accumulated regions
    zero_kernel<<<(P_ROWS * D_DIM + 255) / 256, 256, 0, stream>>>(cooc, usage, ctx);

    // 1) co-occurrence
    {
        int total = B_ROWS * (S_LEN - 1);
        cooc_kernel<<<(total + 255) / 256, 256, 0, stream>>>(phon, cooc);
    }

    // 2) inverse row norms
    {
        int waves = B_ROWS + P_ROWS;                      // one wave32 per row
        norms_kernel<<<(waves * 32 + 255) / 256, 256, 0, stream>>>(hid, pat, invH, invP);
    }

    // 3) WMMA GEMM: grid = (M/64, N/128)
    {
        dim3 grid(B_ROWS / 64, P_ROWS / 128);
        gemm_wmma_kernel<<<grid, 256, 0, stream>>>(hid, pat, invH, invP, sim);
    }

    // 4) argmax + scatter
    argmax_scatter_kernel<<<(B_ROWS * 32 + 255) / 256, 256, 0, stream>>>(sim, hid, top, usage, ctx);
}